// CombinedLoss_781684048617
// MI455X (gfx1250) — compile-verified
//
#include <hip/hip_runtime.h>
#include <cstdint>
#include <cstddef>

// ---------------------------------------------------------------------------
// CombinedLoss for MI455X (gfx1250).
// Memory-bound streaming reduction (252 MB in -> 3 floats out, ~10.8us floor
// at 23.3 TB/s). Uses CDNA5 async global->LDS copies (ASYNCcnt) for a
// per-wave double-buffered staging pipeline, VALU for the per-slot math,
// wave32 shuffle reduction, f64 global atomics, tiny finalize kernel.
// ---------------------------------------------------------------------------

#define ITEM_    24      // 16 logits + 8 params
#define TPB      128     // 4 waves per block
#define WPB      4       // waves per block
#define TILE     32      // slots per wave-tile
#define SLOTPAD  28      // floats per slot in LDS (24 data + 4 pad; 112B keeps
                         // 16B alignment and only 2-way bank conflict)
#define ARR_B    (TILE * SLOTPAD * 4)   // 3584 bytes per array per tile
#define BUF_B    (2 * ARR_B)            // 7168 bytes (pred + true)
#define WAVE_B   (2 * BUF_B)            // 14336 bytes (double buffered)
#define SMEM_F   (WPB * WAVE_B / 4)     // 14336 floats = 57344 bytes

// Stage one 32-slot array tile (32*96B contiguous in global) into LDS with
// per-slot padding. 6 issues/wave, each moving 32 lanes * 16B = 512B coalesced.
__device__ __forceinline__ void stage_arr(const char* g, unsigned ldsByteBase,
                                          int lane) {
#pragma unroll
  for (int i = 0; i < 6; ++i) {
    int c = i * 32 + lane;          // chunk index 0..191 (16B chunks)
    int s = c / 6;                  // slot within tile
    int r = c - s * 6;              // 16B chunk within slot
    unsigned loff = ldsByteBase + (unsigned)(s * (SLOTPAD * 4) + r * 16);
    uint64_t ga = (uint64_t)(uintptr_t)(g + (size_t)c * 16);
    asm volatile("global_load_async_to_lds_b128 %0, %1, off"
                 :: "v"(loff), "v"(ga)
                 : "memory");
  }
}

__device__ __forceinline__ void slot_accum(const float* yp, const float* yt,
                                           const int* npr,
                                           float& aCls, float& aM, float& aP,
                                           float& aSlm, float& aSla) {
  // log_softmax over 16 logits (max-shifted, matches jax.nn.log_softmax)
  float mx = yp[0];
#pragma unroll
  for (int i = 1; i < 16; ++i) mx = fmaxf(mx, yp[i]);
  float se = 0.f, sl = 0.f;
#pragma unroll
  for (int i = 0; i < 16; ++i) {
    se += __expf(yp[i] - mx);
    sl += yp[i];
  }
  float lse = mx + __logf(se);

  // Running first-argmax over the one-hot, selecting the matching logit and
  // per-class param count simultaneously (no dynamic register indexing).
  float tmax = -__builtin_inff();
  float lpsel = yp[0];
  int   npsel = npr[0];
  float tsum = 0.f;
#pragma unroll
  for (int i = 0; i < 16; ++i) {
    float tv = yt[i];
    tsum += tv;
    bool g = tv > tmax;
    tmax  = g ? tv : tmax;
    lpsel = g ? yp[i] : lpsel;
    npsel = g ? npr[i] : npsel;
  }
  bool mask = tsum > 0.f;

  // -( (1-LS)*logp_true + LS/C * sum(logp) ); sum(logp) = sum(x) - 16*lse
  float cls = -(0.95f * (lpsel - lse) + 0.003125f * (sl - 16.f * lse));
  if (mask) { aCls += cls; aM += 1.f; }

#pragma unroll
  for (int j = 0; j < 8; ++j) {
    float d  = yp[16 + j] - yt[16 + j];
    float ad = fabsf(d);
    float s1 = (ad < 1.f) ? (0.5f * d * d) : (ad - 0.5f);
    aSla += s1;                            // unmasked sum over all (B,E,P)
    if (mask && (j < npsel)) { aSlm += s1; aP += 1.f; }
  }
}

__global__ __launch_bounds__(TPB) void combined_loss_main(
    const float* __restrict__ yp, const float* __restrict__ yt,
    const int* __restrict__ npp, double* __restrict__ acc, int n_slots) {
  __shared__ float smem[SMEM_F];

  const int lane = threadIdx.x & 31;
  const int wib  = threadIdx.x >> 5;
  const long long gw = (long long)blockIdx.x * WPB + wib;   // global wave id
  const long long nw = (long long)gridDim.x * WPB;          // total waves
  const long long ntiles = n_slots >> 5;                    // full 32-slot tiles

  int npr[16];
#pragma unroll
  for (int k = 0; k < 16; ++k) npr[k] = npp[k];

  float aCls = 0.f, aM = 0.f, aP = 0.f, aSlm = 0.f, aSla = 0.f;

  const unsigned smemBase = (unsigned)(uintptr_t)(&smem[0]); // LDS byte offset
  const unsigned waveByte = smemBase + (unsigned)(wib * WAVE_B);

  long long t = gw;
  int buf = 0;
  if (t < ntiles) {   // prologue: stage first tile (wave-uniform branch)
    stage_arr((const char*)(yp + t * (TILE * ITEM_)), waveByte, lane);
    stage_arr((const char*)(yt + t * (TILE * ITEM_)), waveByte + ARR_B, lane);
  }
  while (t < ntiles) {
    long long nx = t + nw;
    if (nx < ntiles) {
      // Close the WAR window: async writes into the recycled buffer are not
      // ordered against this wave's in-flight DS reads, so make sure all
      // prior LDS reads have retired (free: compute already consumed them).
      asm volatile("s_wait_dscnt 0x0" ::: "memory");
      unsigned nb = waveByte + (unsigned)((buf ^ 1) * BUF_B);
      stage_arr((const char*)(yp + nx * (TILE * ITEM_)), nb, lane);
      stage_arr((const char*)(yt + nx * (TILE * ITEM_)), nb + ARR_B, lane);
      // 12 newer async ops in flight; <=12 means current tile's 12 are done.
      asm volatile("s_wait_asynccnt 0xc" ::: "memory");
    } else {
      asm volatile("s_wait_asynccnt 0x0" ::: "memory");
    }

    const int bufF = wib * (WAVE_B / 4) + buf * (BUF_B / 4);
    const float* pp = &smem[bufF + lane * SLOTPAD];
    const float* tp = &smem[bufF + (ARR_B / 4) + lane * SLOTPAD];
    float A[24], Bv[24];
#pragma unroll
    for (int k = 0; k < 24; ++k) { A[k] = pp[k]; Bv[k] = tp[k]; }
    slot_accum(A, Bv, npr, aCls, aM, aP, aSlm, aSla);

    buf ^= 1;
    t = nx;
  }

  // Tail slots (n_slots % 32), direct global loads.
  {
    long long base = ntiles * 32;
    long long gtid = (long long)blockIdx.x * blockDim.x + threadIdx.x;
    long long s = base + gtid;
    if (s < n_slots) {
      const float* p = yp + s * ITEM_;
      const float* q = yt + s * ITEM_;
      float A[24], Bv[24];
#pragma unroll
      for (int k = 0; k < 24; ++k) { A[k] = p[k]; Bv[k] = q[k]; }
      slot_accum(A, Bv, npr, aCls, aM, aP, aSlm, aSla);
    }
  }

  // wave32 butterfly reduction
#pragma unroll
  for (int off = 16; off > 0; off >>= 1) {
    aCls += __shfl_xor(aCls, off);
    aM   += __shfl_xor(aM,   off);
    aP   += __shfl_xor(aP,   off);
    aSlm += __shfl_xor(aSlm, off);
    aSla += __shfl_xor(aSla, off);
  }
  if (lane == 0) {
    atomicAdd(&acc[0], (double)aCls);
    atomicAdd(&acc[1], (double)aM);
    atomicAdd(&acc[2], (double)aP);
    atomicAdd(&acc[3], (double)aSlm);
    atomicAdd(&acc[4], (double)aSla);
  }
}

__global__ void combined_loss_zero(double* acc) {
  if (threadIdx.x < 5) acc[threadIdx.x] = 0.0;
}

__global__ void combined_loss_finalize(const double* __restrict__ acc,
                                       float* __restrict__ out) {
  double sc  = acc[0];   // sum(cls * mask)
  double ms  = acc[1];   // msum
  double ps  = acc[2];   // psum
  double slm = acc[3];   // sum(sl1 * param_mask)
  double sla = acc[4];   // sum(sl1) over everything
  double loss_cls = (ms > 0.0) ? (sc / fmax(ms, 1.0)) : 0.0;
  double reg_m = slm / fmax(ps, 1.0);
  double reg_u = sla / fmax(ms, 1.0);
  double loss_reg = (ms > 0.0) ? ((ps > 0.0) ? reg_m : reg_u) : 0.0;
  out[0] = (float)(loss_cls + loss_reg);   // total (REG_W = 1.0)
  out[1] = (float)loss_cls;
  out[2] = (float)loss_reg;
}

extern "C" void kernel_launch(void* const* d_in, const int* in_sizes, int n_in,
                              void* d_out, int out_size, void* d_ws,
                              size_t ws_size, hipStream_t stream) {
  const float* yp  = (const float*)d_in[0];
  const float* yt  = (const float*)d_in[1];
  const int*   npp = (const int*)d_in[2];
  float*  out = (float*)d_out;
  double* acc = (double*)d_ws;          // 5 doubles of scratch

  const int n_slots = in_sizes[0] / ITEM_;   // B * E

  combined_loss_zero<<<1, 32, 0, stream>>>(acc);
  combined_loss_main<<<1024, TPB, 0, stream>>>(yp, yt, npp, acc, n_slots);
  combined_loss_finalize<<<1, 1, 0, stream>>>(acc, out);
}